// GCNEncoder_57071525429450
// MI455X (gfx1250) — compile-verified
//
#include <hip/hip_runtime.h>
#include <math.h>

// fp32 WMMA operand types: A/B of 16x16x4_f32 are 2 VGPRs, C/D are 8 VGPRs.
typedef __attribute__((ext_vector_type(2))) float v2f;
typedef __attribute__((ext_vector_type(8))) float v8f;

// Native agent-scope fp32 atomic add -> global_atomic_add_f32 (no CAS loop).
__device__ __forceinline__ void atomic_add_f32(float* p, float v) {
    __hip_atomic_fetch_add(p, v, __ATOMIC_RELAXED, __HIP_MEMORY_SCOPE_AGENT);
}

// ---------------------------------------------------------------- degree prep
__global__ void k_fill_deg(float* __restrict__ deg, int n) {
    int i = blockIdx.x * blockDim.x + threadIdx.x;
    if (i < n) deg[i] = 1.0f;  // self-loop contributes 1
}

__global__ void k_deg_scatter(const int* __restrict__ dst, int E,
                              float* __restrict__ deg) {
    int e = blockIdx.x * blockDim.x + threadIdx.x;
    if (e < E) atomic_add_f32(&deg[dst[e]], 1.0f);
}

__global__ void k_dinv(float* __restrict__ deg, int n) {
    int i = blockIdx.x * blockDim.x + threadIdx.x;
    if (i < n) {
        float d = deg[i];
        deg[i] = (d > 0.0f) ? rsqrtf(d) : 0.0f;
    }
}

// ------------------------------------------------------------- fp32 WMMA GEMM
// C[M, NT*16] = op(A[M, KD]) @ B[KD, NT*16], op = ReLU if RELU.
// Block = 256 threads = 8 waves; each wave owns one 16-row M-tile and sweeps
// all NT column tiles, so each A fragment load feeds NT WMMAs. B is staged in
// LDS transposed ([col][k], stride KD+2 for bank-conflict-free ds_load_b64).
// Requires M % 16 == 0 (50000 = 3125*16); KD, NT*16 compile-time.
template <int NT, int KD, bool RELU>
__global__ void __launch_bounds__(256) k_gemm_wmma_f32(
    const float* __restrict__ A, const float* __restrict__ B,
    float* __restrict__ C, int M)
{
    constexpr int N   = NT * 16;
    constexpr int LDK = KD + 2;                 // padded K-stride in LDS
    __shared__ float lds_w[N * LDK];

    // Cooperative load of W, coalesced read, transposed write.
    for (int idx = threadIdx.x; idx < KD * N; idx += 256) {
        int k = idx / N;                        // compile-time N -> shifts
        int n = idx - k * N;
        lds_w[n * LDK + k] = B[idx];
    }
    __syncthreads();

    const int lane  = threadIdx.x & 31;
    const int wv    = threadIdx.x >> 5;         // 0..7
    const int mtile = blockIdx.x * 8 + wv;      // wave-uniform
    if (mtile * 16 >= M) return;                // whole-wave exit, EXEC all-1s

    const int hf = lane >> 4;                   // selects K pair (0: k,k+1 / 1: k+2,k+3)
    const int r  = lane & 15;                   // row (A) / col (B,C) in tile

    const float* __restrict__ arow = A + (size_t)(mtile * 16 + r) * KD + 2 * hf;

    v8f acc[NT];
#pragma unroll
    for (int t = 0; t < NT; ++t) acc[t] = (v8f){};

#pragma unroll 4
    for (int k = 0; k < KD; k += 4) {
        float2 av = *(const float2*)(arow + k);          // 8B global load
        if (RELU) { av.x = fmaxf(av.x, 0.0f); av.y = fmaxf(av.y, 0.0f); }
        v2f a;
        a[0] = av.x;
        a[1] = av.y;
#pragma unroll
        for (int t = 0; t < NT; ++t) {
            const int col = t * 16 + r;
            float2 bv = *(const float2*)(&lds_w[col * LDK + k + 2 * hf]); // ds_load_b64
            v2f b;
            b[0] = bv.x;
            b[1] = bv.y;
            // (neg_a, A, neg_b, B, c_mod, C, reuse_a, reuse_b)
            acc[t] = __builtin_amdgcn_wmma_f32_16x16x4_f32(
                false, a, false, b, (short)0, acc[t], false, false);
        }
    }

    // C/D layout: lanes 0-15 -> M = mtile*16 + j, lanes 16-31 -> +8.
#pragma unroll
    for (int t = 0; t < NT; ++t) {
        float* crow = C + (size_t)(mtile * 16 + 8 * hf) * N + t * 16 + r;
#pragma unroll
        for (int j = 0; j < 8; ++j) crow[(size_t)j * N] = acc[t][j];
    }
}

// ---------------------------------------- fused bias + self-loop init of dest
// out[i,c] = bias[c] + h[i,c] * dinv[i]^2 ; C = 1<<lgC (128 or 64).
__global__ void k_self_bias_init(const float* __restrict__ h,
                                 const float* __restrict__ dinv,
                                 const float* __restrict__ bias,
                                 float* __restrict__ out, int n, int lgC) {
    long long t = (long long)blockIdx.x * blockDim.x + threadIdx.x;
    long long total = (long long)n << lgC;
    if (t >= total) return;
    int i = (int)(t >> lgC);
    int c = (int)(t & (((long long)1 << lgC) - 1));
    float di = dinv[i];
    out[t] = bias[c] + h[t] * di * di;
}

// ------------------------------------------------- edge gather-scale-scatter
// out[dst,c] += h[src,c] * dinv[src]*dinv[dst]; float4 gather + f32 atomics.
// Channel groups per edge = 1<<lgG (32 or 16); C = 4<<lgG.
__global__ void k_edge_scatter(const int* __restrict__ src,
                               const int* __restrict__ dst,
                               const float* __restrict__ dinv,
                               const float* __restrict__ h,
                               float* __restrict__ out, int E, int lgG) {
    long long t = (long long)blockIdx.x * blockDim.x + threadIdx.x;
    long long total = (long long)E << lgG;
    if (t >= total) return;
    const int C = 4 << lgG;
    int e  = (int)(t >> lgG);
    int cg = (int)(t & (((long long)1 << lgG) - 1)) << 2;
    int s = src[e], d = dst[e];
    float nrm = dinv[s] * dinv[d];
    const float4 hv = *(const float4*)(h + (size_t)s * C + cg);
    float* o = out + (size_t)d * C + cg;
    atomic_add_f32(o + 0, hv.x * nrm);
    atomic_add_f32(o + 1, hv.y * nrm);
    atomic_add_f32(o + 2, hv.z * nrm);
    atomic_add_f32(o + 3, hv.w * nrm);
}

// ---------------------------------------------------------------------- host
extern "C" void kernel_launch(void* const* d_in, const int* in_sizes, int n_in,
                              void* d_out, int out_size, void* d_ws, size_t ws_size,
                              hipStream_t stream) {
    const float* x  = (const float*)d_in[0];
    const int*   ei = (const int*)d_in[1];   // JAX canonicalizes to int32
    const float* W1 = (const float*)d_in[2];
    const float* b1 = (const float*)d_in[3];
    const float* W2 = (const float*)d_in[4];
    const float* b2 = (const float*)d_in[5];
    float* out = (float*)d_out;

    const int HID = in_sizes[3];            // 128
    const int OUT = in_sizes[5];            // 64
    const int IN  = in_sizes[2] / HID;      // 128
    const int N   = in_sizes[0] / IN;       // 50000 (multiple of 16)
    const int E   = in_sizes[1] / 2;        // 800000
    const int* srcI = ei;
    const int* dstI = ei + E;

    // Workspace: dinv[N] | bufA[N*HID] | bufB[N*HID]  (~51.4 MB, L2-resident)
    float* ws   = (float*)d_ws;
    float* dinv = ws;
    float* bufA = ws + (((size_t)N + 63) & ~(size_t)63);
    float* bufB = bufA + (size_t)N * HID;

    const int T = 256;
    const int mtiles = N / 16;                   // 3125
    const int gblk   = (mtiles + 7) / 8;         // 8 waves (M-tiles) per block

    // --- degrees / normalization ---
    k_fill_deg   <<<(N + T - 1) / T, T, 0, stream>>>(dinv, N);
    k_deg_scatter<<<(E + T - 1) / T, T, 0, stream>>>(dstI, E, dinv);
    k_dinv       <<<(N + T - 1) / T, T, 0, stream>>>(dinv, N);

    // --- layer 1: h1 = x @ W1  (NT=8 -> N=128, KD=128) ---
    k_gemm_wmma_f32<8, 128, false><<<gblk, T, 0, stream>>>(x, W1, bufA, N);
    // agg1 = b1 + h1*dinv^2 (self-loop + bias), then edge scatter-add
    {
        long long tot = (long long)N * HID;      // lgC = 7
        k_self_bias_init<<<(int)((tot + T - 1) / T), T, 0, stream>>>(
            bufA, dinv, b1, bufB, N, 7);
        long long etot = (long long)E << 5;      // lgG = 5 (128/4 groups)
        k_edge_scatter<<<(int)((etot + T - 1) / T), T, 0, stream>>>(
            srcI, dstI, dinv, bufA, bufB, E, 5);
    }

    // --- layer 2: h2 = relu(agg1) @ W2  (NT=4 -> N=64, ReLU fused in A load) ---
    k_gemm_wmma_f32<4, 128, true><<<gblk, T, 0, stream>>>(bufB, W2, bufA, N);
    // d_out = b2 + h2*dinv^2, then edge scatter-add into d_out
    {
        long long tot = (long long)N * OUT;      // lgC = 6
        k_self_bias_init<<<(int)((tot + T - 1) / T), T, 0, stream>>>(
            bufA, dinv, b2, out, N, 6);
        long long etot = (long long)E << 4;      // lgG = 4 (64/4 groups)
        k_edge_scatter<<<(int)((etot + T - 1) / T), T, 0, stream>>>(
            srcI, dstI, dinv, bufA, out, E, 4);
    }
}